// AttnRNNEncoder_13469017440704
// MI455X (gfx1250) — compile-verified
//
#include <hip/hip_runtime.h>
#include <hip/hip_bf16.h>
#include <math.h>

// RNN encoder: x_proj = gather(W_emb, src) @ W_xh^T + b_xh ; then scan of
// h_t = tanh(xp_t + h_{t-1} @ W_hh^T). All f32 -> V_WMMA_F32_16X16X4_F32,
// register-blocked so the WMMA pipe (not cache bandwidth) is the limiter.

typedef __attribute__((ext_vector_type(2))) float v2f;
typedef __attribute__((ext_vector_type(8))) float v8f;

#define BATCH 64
#define SEQ   512
#define EMB   1024
#define HID   1024
#define MTOT  (BATCH * SEQ)   // 32768 rows of the batched input GEMM

#define WMMA_F32(A, B, C)                                              \
  __builtin_amdgcn_wmma_f32_16x16x4_f32(false, (A), false, (B),        \
                                        (short)0, (C), false, false)

// ---------------------------------------------------------------------------
// Kernel 1: x_proj[m, n] = sum_k W_emb[src[m], k] * W_xh[n, k] + b_xh[n]
// One wave computes a 32x64 output tile: 2 A-fragments x 4 B-fragments ->
// 8 f32 accumulators; 6 b64 loads feed 8 WMMAs per K=4 step (~10.7 FLOP/B).
// A fragment (16x4 f32, ISA 7.12.2): lanes 0-15 hold K={0,1}, lanes 16-31
// hold K={2,3}, row M = lane%16. B fragment (4x16) mirrors it per column.
// ---------------------------------------------------------------------------
__global__ __launch_bounds__(256) void xproj_wmma(
    const int* __restrict__ src, const float* __restrict__ Wemb,
    const float* __restrict__ Wxh, const float* __restrict__ bxh,
    float* __restrict__ out)
{
  const int lane  = threadIdx.x & 31;
  const int wave  = threadIdx.x >> 5;
  const int mtile = blockIdx.x >> 1;                   // 1024 M-supertiles (32 rows)
  const int nblk  = ((blockIdx.x & 1) << 3) + wave;    // 16 N-supertiles (64 cols)
  const int m0 = mtile << 5;
  const int n0 = nblk  << 6;

  const int lhalf = lane >> 4;      // 0: K pair {0,1}, 1: K pair {2,3}
  const int l16   = lane & 15;
  const int koff  = lhalf << 1;

  // Embedding gather for the two A row-groups of this wave.
  const float* __restrict__ arow0 = Wemb + (size_t)src[m0 + l16]      * EMB;
  const float* __restrict__ arow1 = Wemb + (size_t)src[m0 + 16 + l16] * EMB;
  // B = W_xh^T: column n of B tile j is row (n0 + 16j + lane%16) of W_xh.
  const float* __restrict__ brow0 = Wxh + (size_t)(n0 + l16) * EMB;
  const float* __restrict__ brow1 = brow0 + (size_t)16 * EMB;
  const float* __restrict__ brow2 = brow0 + (size_t)32 * EMB;
  const float* __restrict__ brow3 = brow0 + (size_t)48 * EMB;

  v8f c00 = {}, c01 = {}, c02 = {}, c03 = {};
  v8f c10 = {}, c11 = {}, c12 = {}, c13 = {};

  for (int kc = 0; kc < EMB; kc += 128) {
    if (kc + 128 < EMB) {   // stream next 512B chunk of each row ahead
      __builtin_prefetch(arow0 + kc + 128 + koff, 0, 3);
      __builtin_prefetch(arow1 + kc + 128 + koff, 0, 3);
      __builtin_prefetch(brow0 + kc + 128 + koff, 0, 3);
      __builtin_prefetch(brow1 + kc + 128 + koff, 0, 3);
      __builtin_prefetch(brow2 + kc + 128 + koff, 0, 3);
      __builtin_prefetch(brow3 + kc + 128 + koff, 0, 3);
    }
#pragma unroll 4
    for (int k = kc; k < kc + 128; k += 4) {
      const v2f a0 = *(const v2f*)(arow0 + k + koff);
      const v2f a1 = *(const v2f*)(arow1 + k + koff);
      const v2f b0 = *(const v2f*)(brow0 + k + koff);
      const v2f b1 = *(const v2f*)(brow1 + k + koff);
      const v2f b2 = *(const v2f*)(brow2 + k + koff);
      const v2f b3 = *(const v2f*)(brow3 + k + koff);
      c00 = WMMA_F32(a0, b0, c00);
      c01 = WMMA_F32(a0, b1, c01);
      c02 = WMMA_F32(a0, b2, c02);
      c03 = WMMA_F32(a0, b3, c03);
      c10 = WMMA_F32(a1, b0, c10);
      c11 = WMMA_F32(a1, b1, c11);
      c12 = WMMA_F32(a1, b2, c12);
      c13 = WMMA_F32(a1, b3, c13);
    }
  }

  // C/D layout: VGPR r, lanes 0-15 -> M = r, lanes 16-31 -> M = r + 8.
  const float bias0 = bxh[n0 + l16];
  const float bias1 = bxh[n0 + 16 + l16];
  const float bias2 = bxh[n0 + 32 + l16];
  const float bias3 = bxh[n0 + 48 + l16];

  float* cbase0 = out + (size_t)(m0 + (lhalf << 3)) * HID + n0 + l16;       // A-group 0
  float* cbase1 = out + (size_t)(m0 + 16 + (lhalf << 3)) * HID + n0 + l16;  // A-group 1
#pragma unroll
  for (int r = 0; r < 8; ++r) {
    const size_t ro = (size_t)r * HID;
    cbase0[ro]      = c00[r] + bias0;
    cbase0[ro + 16] = c01[r] + bias1;
    cbase0[ro + 32] = c02[r] + bias2;
    cbase0[ro + 48] = c03[r] + bias3;
    cbase1[ro]      = c10[r] + bias0;
    cbase1[ro + 16] = c11[r] + bias1;
    cbase1[ro + 32] = c12[r] + bias2;
    cbase1[ro + 48] = c13[r] + bias3;
  }
}

// ---------------------------------------------------------------------------
// Kernel 2 (launched once per timestep t): in-place update of out[:, t, :]:
//   out[b, t, n] = tanh(out[b, t, n] + sum_k out[b, t-1, k] * W_hh[n, k])
// One wave per 32x32 tile: 2x2 fragments, 4 loads feed 4 WMMAs per K-step.
// 64 waves over 8 blocks keep the serial step spread across WGPs.
// ---------------------------------------------------------------------------
__global__ __launch_bounds__(256) void rnn_step_wmma(
    const float* __restrict__ Whh, float* __restrict__ out, int t)
{
  const int lane  = threadIdx.x & 31;
  const int wave  = threadIdx.x >> 5;
  const int tile  = (blockIdx.x << 3) + wave;   // 0..63
  const int mtile = tile >> 5;                  // 0..1  (batch supertiles of 32)
  const int ntile = tile & 31;                  // 0..31 (hidden supertiles of 32)
  const int m0 = mtile << 5;
  const int n0 = ntile << 5;

  const int lhalf = lane >> 4;
  const int l16   = lane & 15;
  const int koff  = lhalf << 1;

  v8f c00 = {}, c01 = {}, c10 = {}, c11 = {};

  if (t > 0) {  // uniform branch: EXEC stays all-ones around the WMMAs
    const float* __restrict__ arow0 =
        out + (size_t)(m0 + l16) * (SEQ * HID) + (size_t)(t - 1) * HID;
    const float* __restrict__ arow1 = arow0 + (size_t)16 * (SEQ * HID);
    const float* __restrict__ brow0 = Whh + (size_t)(n0 + l16) * HID;
    const float* __restrict__ brow1 = brow0 + (size_t)16 * HID;

    for (int kc = 0; kc < HID; kc += 128) {
      if (kc + 128 < HID) {
        __builtin_prefetch(arow0 + kc + 128 + koff, 0, 3);
        __builtin_prefetch(arow1 + kc + 128 + koff, 0, 3);
        __builtin_prefetch(brow0 + kc + 128 + koff, 0, 3);
        __builtin_prefetch(brow1 + kc + 128 + koff, 0, 3);
      }
#pragma unroll 4
      for (int k = kc; k < kc + 128; k += 4) {
        const v2f a0 = *(const v2f*)(arow0 + k + koff);
        const v2f a1 = *(const v2f*)(arow1 + k + koff);
        const v2f b0 = *(const v2f*)(brow0 + k + koff);
        const v2f b1 = *(const v2f*)(brow1 + k + koff);
        c00 = WMMA_F32(a0, b0, c00);
        c01 = WMMA_F32(a0, b1, c01);
        c10 = WMMA_F32(a1, b0, c10);
        c11 = WMMA_F32(a1, b1, c11);
      }
    }
  }

  float* cbase0 = out + (size_t)(m0 + (lhalf << 3)) * (SEQ * HID)
                      + (size_t)t * HID + n0 + l16;
  float* cbase1 = cbase0 + (size_t)16 * (SEQ * HID);
#pragma unroll
  for (int r = 0; r < 8; ++r) {
    const size_t ro = (size_t)r * (SEQ * HID);
    cbase0[ro]      = tanhf(cbase0[ro]      + c00[r]);
    cbase0[ro + 16] = tanhf(cbase0[ro + 16] + c01[r]);
    cbase1[ro]      = tanhf(cbase1[ro]      + c10[r]);
    cbase1[ro + 16] = tanhf(cbase1[ro + 16] + c11[r]);
  }
}

// ---------------------------------------------------------------------------
// Kernel 3: h_last[b, h] = out[b, S-1, h]
// ---------------------------------------------------------------------------
__global__ __launch_bounds__(256) void copy_hlast(
    const float* __restrict__ out, float* __restrict__ hlast)
{
  const int i = blockIdx.x * blockDim.x + threadIdx.x;  // 65536 elements
  const int b = i >> 10;
  const int h = i & 1023;
  hlast[i] = out[(size_t)b * (SEQ * HID) + (size_t)(SEQ - 1) * HID + h];
}

extern "C" void kernel_launch(void* const* d_in, const int* in_sizes, int n_in,
                              void* d_out, int out_size, void* d_ws, size_t ws_size,
                              hipStream_t stream) {
  (void)in_sizes; (void)n_in; (void)out_size; (void)d_ws; (void)ws_size;
  const int*   src  = (const int*)d_in[0];    // [64, 512] int32
  const float* Wemb = (const float*)d_in[1];  // [32000, 1024]
  const float* Wxh  = (const float*)d_in[2];  // [1024, 1024]
  const float* bxh  = (const float*)d_in[3];  // [1024]
  const float* Whh  = (const float*)d_in[4];  // [1024, 1024]

  float* out   = (float*)d_out;                       // [64, 512, 1024]
  float* hlast = out + (size_t)MTOT * HID;            // [64, 1024]

  // Batched input projection (writes x_proj into the outputs buffer).
  // 1024 M-supertiles x 2 blocks-per-supertile = 2048 blocks of 8 waves.
  xproj_wmma<<<2048, 256, 0, stream>>>(src, Wemb, Wxh, bxh, out);

  // Sequential recurrence: one small WMMA GEMM + fused tanh per timestep.
  for (int t = 0; t < SEQ; ++t) {
    rnn_step_wmma<<<8, 256, 0, stream>>>(Whh, out, t);
  }

  // Final hidden state.
  copy_hlast<<<256, 256, 0, stream>>>(out, hlast);
}